// TorchSNNext_86440511799558
// MI455X (gfx1250) — compile-verified
//
#include <hip/hip_runtime.h>
#include <hip/hip_bf16.h>

// ---------------------------------------------------------------------------
// Constants from the reference (decay factors folded to f32 constants)
// ---------------------------------------------------------------------------
#define SOMA_GL   1e-8f
#define SOMA_EL  -0.070f
#define AIS_GL    1e-8f
#define AIS_EL   -0.070f
#define V_T      -0.050f
#define DELTA_T   0.002f
#define V_SPIKE  -0.030f
#define V_RESET  -0.065f
#define G_C       5e-9f
#define G_MAX     1e-9f
#define A_PLUS    0.01f
#define A_MINUS   0.012f
#define ETA_BCM   1e-4f
#define DT_OVER_SOMA_C 5.0e5f      // 1e-4 / 2e-10
#define DT_OVER_AIS_C  1.0e6f      // 1e-4 / 1e-10
#define DECAY_G    0.980198673306755f   // exp(-1e-4/0.005)
#define DECAY_PRE  0.995012479192682f   // exp(-1e-4/0.020)
#define DECAY_RATE 0.999000499833375f   // exp(-1e-4/0.100)
#define DT_OVER_TAU_THETA 1e-4f

#define THREADS   256
#define VEC       4
#define TILE_SYN  (THREADS * VEC)   // 1024 synapses / block; LDS tile = 4 KB

// ---------------------------------------------------------------------------
// TDM availability probe (this toolchain: 6-arg builtin)
// ---------------------------------------------------------------------------
#if defined(__has_builtin)
#if __has_builtin(__builtin_amdgcn_tensor_load_to_lds) && __has_builtin(__builtin_amdgcn_s_wait_tensorcnt)
#define SNN_USE_TDM 1
#endif
#endif
#ifndef SNN_USE_TDM
#define SNN_USE_TDM 0
#endif

typedef __attribute__((ext_vector_type(4))) unsigned int u32x4;
typedef __attribute__((ext_vector_type(8))) int          i32x8;
typedef __attribute__((ext_vector_type(4))) int          i32x4;

// ---------------------------------------------------------------------------
// K0: zero the per-neuron synaptic-current accumulator (workspace)
// ---------------------------------------------------------------------------
__global__ void snn_zero_kernel(float* __restrict__ p, int n) {
    int i = blockIdx.x * blockDim.x + threadIdx.x;
    if (i < n) p[i] = 0.0f;
}

// ---------------------------------------------------------------------------
// K1: i_syn = g*w*G_MAX*(0 - V_soma[post]);  segment_sum via FP32 atomics.
//     float4/int4 streaming loads; gathers + atomics remain scalar.
// ---------------------------------------------------------------------------
__global__ void __launch_bounds__(THREADS)
snn_syn_current_kernel(const float* __restrict__ syn_g,
                       const float* __restrict__ syn_w,
                       const int*   __restrict__ post_idx,
                       const float* __restrict__ V_soma,
                       float*       __restrict__ I_syn,
                       int S_) {
    const int i0 = (blockIdx.x * THREADS + threadIdx.x) * VEC;
    if (i0 >= S_) return;
    __builtin_prefetch(&syn_g[i0 + 8 * TILE_SYN], 0, 1);   // global_prefetch_b8
    __builtin_prefetch(&syn_w[i0 + 8 * TILE_SYN], 0, 1);
    if (i0 + VEC <= S_) {
        const float4 g4 = *(const float4*)(syn_g + i0);    // global_load_b128
        const float4 w4 = *(const float4*)(syn_w + i0);
        const int4   p4 = *(const int4*)(post_idx + i0);
        const float gw[VEC] = {g4.x * w4.x, g4.y * w4.y, g4.z * w4.z, g4.w * w4.w};
        const int   pp[VEC] = {p4.x, p4.y, p4.z, p4.w};
        #pragma unroll
        for (int j = 0; j < VEC; ++j) {
            float isyn = gw[j] * G_MAX * (0.0f - V_soma[pp[j]]);
            __hip_atomic_fetch_add(&I_syn[pp[j]], isyn,
                                   __ATOMIC_RELAXED, __HIP_MEMORY_SCOPE_AGENT);
        }
    } else {
        for (int i = i0; i < S_; ++i) {
            int   p    = post_idx[i];
            float isyn = syn_g[i] * syn_w[i] * G_MAX * (0.0f - V_soma[p]);
            __hip_atomic_fetch_add(&I_syn[p], isyn,
                                   __ATOMIC_RELAXED, __HIP_MEMORY_SCOPE_AGENT);
        }
    }
}

// ---------------------------------------------------------------------------
// K2: two-compartment neuron step (float4); forward spike value == hard
// ---------------------------------------------------------------------------
__device__ __forceinline__ void neuron_step(float vs, float va, float iext,
                                            float isyn, float refrac,
                                            float& spk, float& vs_o, float& va_o) {
    float dvs = DT_OVER_SOMA_C * (-SOMA_GL * (vs - SOMA_EL) + iext
                                  + isyn + G_C * (va - vs));
    float e   = AIS_GL * DELTA_T * __expf(fminf((va - V_T) / DELTA_T, 20.0f));
    float dva = DT_OVER_AIS_C * (-AIS_GL * (va - AIS_EL) + e + G_C * (vs - va));
    float va_new = (0.0f < refrac) ? V_RESET : (va + dva);   // T_NOW = 0
    float hard   = (va_new >= V_SPIKE) ? 1.0f : 0.0f;        // forward spike value
    spk  = hard;
    vs_o = vs + dvs;
    va_o = (hard > 0.0f) ? V_RESET : va_new;
}

__global__ void __launch_bounds__(THREADS)
snn_neuron_kernel(const float* __restrict__ ext_cur,
                  const float* __restrict__ V_soma,
                  const float* __restrict__ V_ais,
                  const float* __restrict__ refrac_until,
                  const float* __restrict__ I_syn,
                  float* __restrict__ o_spikes,
                  float* __restrict__ o_vsoma,
                  float* __restrict__ o_vais,
                  int n) {
    const int i0 = (blockIdx.x * THREADS + threadIdx.x) * VEC;
    if (i0 >= n) return;
    if (i0 + VEC <= n) {
        const float4 vs4 = *(const float4*)(V_soma + i0);
        const float4 va4 = *(const float4*)(V_ais + i0);
        const float4 ex4 = *(const float4*)(ext_cur + i0);
        const float4 rf4 = *(const float4*)(refrac_until + i0);
        const float4 is4 = *(const float4*)(I_syn + i0);
        float4 spk, vso, vao;
        neuron_step(vs4.x, va4.x, ex4.x, is4.x, rf4.x, spk.x, vso.x, vao.x);
        neuron_step(vs4.y, va4.y, ex4.y, is4.y, rf4.y, spk.y, vso.y, vao.y);
        neuron_step(vs4.z, va4.z, ex4.z, is4.z, rf4.z, spk.z, vso.z, vao.z);
        neuron_step(vs4.w, va4.w, ex4.w, is4.w, rf4.w, spk.w, vso.w, vao.w);
        *(float4*)(o_spikes + i0) = spk;                   // global_store_b128
        *(float4*)(o_vsoma + i0)  = vso;
        *(float4*)(o_vais + i0)   = vao;
    } else {
        for (int i = i0; i < n; ++i)
            neuron_step(V_soma[i], V_ais[i], ext_cur[i], I_syn[i], refrac_until[i],
                        o_spikes[i], o_vsoma[i], o_vais[i]);
    }
}

// ---------------------------------------------------------------------------
// K3: plasticity + conductance + ring-buffer rewrite (float4 per thread).
//     TDM stages the block's contiguous `arrivals` slice (4 KB) into LDS.
// ---------------------------------------------------------------------------
__global__ void __launch_bounds__(THREADS)
snn_syn_update_kernel(const float* __restrict__ syn_g,
                      const float* __restrict__ syn_w,
                      const float* __restrict__ pre_trace,
                      const float* __restrict__ post_trace,
                      const float* __restrict__ r_pre,
                      const float* __restrict__ r_post,
                      const float* __restrict__ theta_m,
                      const float* __restrict__ gbuf,       // L x S
                      const int*   __restrict__ pre_idx,
                      const int*   __restrict__ post_idx,
                      const int*   __restrict__ delay_steps,
                      const int*   __restrict__ buffer_idx_p,
                      const float* __restrict__ spikes,
                      float* __restrict__ o_g,
                      float* __restrict__ o_w,
                      float* __restrict__ o_buf,            // L x S
                      int S_, int L_) {
    __shared__ float s_arr[TILE_SYN];
    const int base = blockIdx.x * TILE_SYN;
    const int bidx = *buffer_idx_p;                    // uniform scalar load
    const float* __restrict__ grow = gbuf + (size_t)bidx * (size_t)S_ + base;
    const bool full_tile = (base + TILE_SYN <= S_);

#if SNN_USE_TDM
    if (full_tile) {
        if (threadIdx.x < 32u) {   // wave 0 issues the DMA (wave-uniform branch)
            unsigned long long ga = (unsigned long long)(uintptr_t)grow;
            unsigned int lds = (unsigned int)(unsigned long long)(uintptr_t)&s_arr[0];
            // D# group0: count=1 | lds_addr | global_addr[56:0] | type=2
            u32x4 g0;
            g0.x = 1u;
            g0.y = lds;
            g0.z = (unsigned int)ga;
            g0.w = (unsigned int)((ga >> 32) & 0x01FFFFFFu) | (2u << 30);
            // D# group1: data_size=4B; 1-D tensor: dim0=TILE_SYN, dim1=1,
            // tile_dim0=TILE_SYN, tile_dim1=1, strides=TILE_SYN
            i32x8 g1 = {
                (int)(2u << 16),                                   // data_size=2 (4B), wg_mask=0
                (int)(((unsigned)TILE_SYN & 0xFFFFu) << 16),       // tensor_dim0[15:0]
                (int)((((unsigned)TILE_SYN >> 16) & 0xFFFFu) | (1u << 16)), // dim0 hi | tensor_dim1=1
                (int)(((unsigned)TILE_SYN & 0xFFFFu) << 16),       // tile_dim0
                1,                                                 // tile_dim1=1, tile_dim2=0
                (int)TILE_SYN,                                     // tensor_dim0_stride lo32
                (int)(((unsigned)TILE_SYN & 0xFFFFu) << 16),       // tensor_dim1_stride lo16
                0
            };
            i32x4 z4 = {0, 0, 0, 0};
            i32x8 z8 = {0, 0, 0, 0, 0, 0, 0, 0};
            __builtin_amdgcn_tensor_load_to_lds(g0, g1, z4, z4, z8, 0);
            __builtin_amdgcn_s_wait_tensorcnt(0);                  // s_wait_tensorcnt 0
        }
    } else {
        for (int j = threadIdx.x; base + j < S_ && j < TILE_SYN; j += THREADS)
            s_arr[j] = grow[j];
    }
#else
    for (int j = threadIdx.x; base + j < S_ && j < TILE_SYN; j += THREADS)
        s_arr[j] = grow[j];
#endif
    __syncthreads();

    const int i0 = base + threadIdx.x * VEC;
    if (i0 >= S_) return;

    __builtin_prefetch(&syn_g[i0 + 8 * TILE_SYN], 0, 1);
    __builtin_prefetch(&theta_m[i0 + 8 * TILE_SYN], 0, 1);

    if (i0 + VEC <= S_) {
        // ---- vector path: b128 streaming, scalar gathers ----
        const float4 arr = *(const float4*)(&s_arr[threadIdx.x * VEC]);  // ds_load_b128
        const float4 g4  = *(const float4*)(syn_g + i0);
        const float4 w4  = *(const float4*)(syn_w + i0);
        const float4 pt4 = *(const float4*)(pre_trace + i0);
        const float4 qt4 = *(const float4*)(post_trace + i0);
        const float4 rp4 = *(const float4*)(r_pre + i0);
        const float4 rq4 = *(const float4*)(r_post + i0);
        const float4 th4 = *(const float4*)(theta_m + i0);
        const int4   pi4 = *(const int4*)(pre_idx + i0);
        const int4   qi4 = *(const int4*)(post_idx + i0);
        const int4   dl4 = *(const int4*)(delay_steps + i0);

        const float ga[VEC] = {arr.x, arr.y, arr.z, arr.w};
        const float gg[VEC] = {g4.x, g4.y, g4.z, g4.w};
        const float ww[VEC] = {w4.x, w4.y, w4.z, w4.w};
        const float pt[VEC] = {pt4.x, pt4.y, pt4.z, pt4.w};
        const float qt[VEC] = {qt4.x, qt4.y, qt4.z, qt4.w};
        const float rp[VEC] = {rp4.x, rp4.y, rp4.z, rp4.w};
        const float rq[VEC] = {rq4.x, rq4.y, rq4.z, rq4.w};
        const float th[VEC] = {th4.x, th4.y, th4.z, th4.w};
        const int   pi[VEC] = {pi4.x, pi4.y, pi4.z, pi4.w};
        const int   qi[VEC] = {qi4.x, qi4.y, qi4.z, qi4.w};
        const int   dl[VEC] = {dl4.x, dl4.y, dl4.z, dl4.w};

        float ps[VEC], go[VEC], wo[VEC];
        int   ra[VEC];
        #pragma unroll
        for (int j = 0; j < VEC; ++j) {
            ps[j] = spikes[pi[j]];                 // random gathers
            const float qs = spikes[qi[j]];
            go[j] = gg[j] * DECAY_G + ga[j];
            const float ptr_ = pt[j] * DECAY_PRE + ps[j];
            const float qtr_ = qt[j] * DECAY_PRE + qs;
            const float dw_stdp = A_PLUS * ptr_ * qs - A_MINUS * qtr_ * ps[j];
            const float rpn = rp[j] * DECAY_RATE + ps[j];
            const float rqn = rq[j] * DECAY_RATE + qs;
            const float thn = th[j] + DT_OVER_TAU_THETA * (rqn * rqn - th[j]);
            const float dw_bcm = ETA_BCM * rpn * rqn * (rqn - thn);
            wo[j] = fminf(fmaxf(ww[j] + dw_stdp + dw_bcm, 0.0f), 1.0f);
            ra[j] = (bidx + dl[j]) % L_;
        }
        *(float4*)(o_g + i0) = make_float4(go[0], go[1], go[2], go[3]);
        *(float4*)(o_w + i0) = make_float4(wo[0], wo[1], wo[2], wo[3]);

        // Ring-buffer rewrite: row bidx cleared; row (bidx+delay)%L gets +pre_spk.
        #pragma unroll 8
        for (int r = 0; r < L_; ++r) {
            float4 v;
            if (r == bidx) v = make_float4(0.0f, 0.0f, 0.0f, 0.0f);
            else           v = *(const float4*)(gbuf + (size_t)r * (size_t)S_ + i0);
            if (r == ra[0]) v.x += ps[0];
            if (r == ra[1]) v.y += ps[1];
            if (r == ra[2]) v.z += ps[2];
            if (r == ra[3]) v.w += ps[3];
            *(float4*)(o_buf + (size_t)r * (size_t)S_ + i0) = v;
        }
    } else {
        // ---- scalar tail ----
        for (int i = i0; i < S_; ++i) {
            const float ps = spikes[pre_idx[i]];
            const float qs = spikes[post_idx[i]];
            o_g[i] = syn_g[i] * DECAY_G + s_arr[i - base];
            const float ptr_ = pre_trace[i]  * DECAY_PRE + ps;
            const float qtr_ = post_trace[i] * DECAY_PRE + qs;
            const float dw_stdp = A_PLUS * ptr_ * qs - A_MINUS * qtr_ * ps;
            const float rpn = r_pre[i]  * DECAY_RATE + ps;
            const float rqn = r_post[i] * DECAY_RATE + qs;
            const float thn = theta_m[i] + DT_OVER_TAU_THETA * (rqn * rqn - theta_m[i]);
            const float dw_bcm = ETA_BCM * rpn * rqn * (rqn - thn);
            o_w[i] = fminf(fmaxf(syn_w[i] + dw_stdp + dw_bcm, 0.0f), 1.0f);
            const int ra = (bidx + delay_steps[i]) % L_;
            for (int r = 0; r < L_; ++r) {
                float v = (r == bidx) ? 0.0f : gbuf[(size_t)r * (size_t)S_ + i];
                if (r == ra) v += ps;
                o_buf[(size_t)r * (size_t)S_ + i] = v;
            }
        }
    }
}

// ---------------------------------------------------------------------------
// Host launcher
// ---------------------------------------------------------------------------
extern "C" void kernel_launch(void* const* d_in, const int* in_sizes, int n_in,
                              void* d_out, int out_size, void* d_ws, size_t ws_size,
                              hipStream_t stream) {
    const float* ext_cur    = (const float*)d_in[0];
    const float* syn_w      = (const float*)d_in[1];
    const float* V_soma     = (const float*)d_in[2];
    const float* V_ais      = (const float*)d_in[3];
    const float* refrac     = (const float*)d_in[4];
    const float* syn_g      = (const float*)d_in[5];
    const float* pre_trace  = (const float*)d_in[6];
    const float* post_trace = (const float*)d_in[7];
    const float* r_pre      = (const float*)d_in[8];
    const float* r_post     = (const float*)d_in[9];
    const float* theta_m    = (const float*)d_in[10];
    const float* gbuf       = (const float*)d_in[11];
    const int*   pre_idx    = (const int*)d_in[12];
    const int*   post_idx   = (const int*)d_in[13];
    const int*   delay      = (const int*)d_in[14];
    const int*   buffer_idx = (const int*)d_in[15];

    const int N_ = in_sizes[0];          // neurons
    const int S_ = in_sizes[1];          // synapses
    const int L_ = in_sizes[11] / S_;    // ring length (8)

    float* I_syn = (float*)d_ws;         // N floats of scratch

    float* out    = (float*)d_out;       // tuple concatenated flat:
    float* o_spk  = out;                 // [N]  spikes
    float* o_vs   = out + (size_t)N_;    // [N]  V_soma_new
    float* o_va   = out + 2 * (size_t)N_;// [N]  V_ais_out
    float* o_g    = out + 3 * (size_t)N_;// [S]  syn_g_new
    float* o_w    = o_g + (size_t)S_;    // [S]  w_new
    float* o_buf  = o_w + (size_t)S_;    // [L*S] buf_new

    snn_zero_kernel<<<(N_ + THREADS - 1) / THREADS, THREADS, 0, stream>>>(I_syn, N_);
    snn_syn_current_kernel<<<(S_ + TILE_SYN - 1) / TILE_SYN, THREADS, 0, stream>>>(
        syn_g, syn_w, post_idx, V_soma, I_syn, S_);
    snn_neuron_kernel<<<(N_ + TILE_SYN - 1) / TILE_SYN, THREADS, 0, stream>>>(
        ext_cur, V_soma, V_ais, refrac, I_syn, o_spk, o_vs, o_va, N_);
    snn_syn_update_kernel<<<(S_ + TILE_SYN - 1) / TILE_SYN, THREADS, 0, stream>>>(
        syn_g, syn_w, pre_trace, post_trace, r_pre, r_post, theta_m,
        gbuf, pre_idx, post_idx, delay, buffer_idx, o_spk,
        o_g, o_w, o_buf, S_, L_);
}